// Conv1d_79413945303390
// MI455X (gfx1250) — compile-verified
//
#include <hip/hip_runtime.h>

// FIR / depthwise cross-correlation via f32 WMMA band-matrix formulation.
// B=32, C=512, L=4096, KL=128 -> out[16384][3969]
//
// D = A x B per V_WMMA_F32_16X16X4_F32:
//   A[m][kk] = x[rowBase+m][i0 + 4p + kk]        (channels in M)
//   B[kk][n] = w[4p + kk - n]  (zero-padded)     (band of shifted weights)
//   => after p = 0..35:  D[m][n] = out[rowBase+m][i0+n]

typedef __attribute__((ext_vector_type(2))) float v2f;
typedef __attribute__((ext_vector_type(8))) float v8f;

#define B_SZ   32
#define C_SZ   512
#define L_SZ   4096
#define KL_SZ  128
#define OL_SZ  (L_SZ - KL_SZ + 1)   // 3969
#define BC_SZ  (B_SZ * C_SZ)        // 16384
#define NSTEP  36                   // covers k = 4p+kk-n in [0,127] for n in [0,15]

__global__ __launch_bounds__(256)
void Conv1d_fir_wmma_kernel(const float* __restrict__ x,
                            const float* __restrict__ w,
                            float* __restrict__ out,
                            int numColTiles) {
    // Band-weight operands: identical for every tile in the dispatch.
    // ldsB[p][lane] = (B[khalf][n], B[khalf+1][n]) with khalf = 2*(lane>>4), n = lane&15.
    __shared__ v2f ldsB[NSTEP * 32];

    const int tid = threadIdx.x;
    for (int e = tid; e < NSTEP * 32; e += 256) {
        const int p     = e >> 5;
        const int ln    = e & 31;
        const int nn    = ln & 15;
        const int khalf = (ln >> 4) * 2;
        const int k0    = 4 * p + khalf - nn;
        const int k1    = k0 + 1;
        v2f b;
        b.x = (k0 >= 0 && k0 < KL_SZ) ? w[k0] : 0.0f;
        b.y = (k1 >= 0 && k1 < KL_SZ) ? w[k1] : 0.0f;
        ldsB[e] = b;
    }
    __syncthreads();

    const int lane = tid & 31;
    const int wave = tid >> 5;
    const int n    = lane & 15;   // N column within tile / A row (M) for this lane
    const int half = lane >> 4;   // 0: K=0..1, 1: K=2..3

    const int colTile = blockIdx.x * 8 + wave;
    if (colTile >= numColTiles) return;

    const int i0      = colTile * 16;
    const int rowBase = blockIdx.y * 16;

    // A operand source row for this lane (M = lane&15 in both lane halves).
    const float* __restrict__ xrow = x + (size_t)(rowBase + n) * L_SZ;

    v8f acc = (v8f){0.f, 0.f, 0.f, 0.f, 0.f, 0.f, 0.f, 0.f};

    if (i0 + (NSTEP - 1) * 4 + 2 + 1 < L_SZ) {
        // Fast path (248 of 249 tiles): one 64-bit base pointer, all loads
        // become global_load_b64 with immediate offsets (zero per-step VALU).
        const float* __restrict__ pA = xrow + i0 + 2 * half;
        #pragma unroll
        for (int p = 0; p < NSTEP; ++p) {
            const v2f a = *(const v2f*)(pA + 4 * p);
            const v2f b = ldsB[p * 32 + lane];
            acc = __builtin_amdgcn_wmma_f32_16x16x4_f32(
                      false, a, false, b, (short)0, acc, false, false);
        }
    } else {
        // Edge path (last column tile only): clamp loads inside the row.
        // Clamped data only ever reaches discarded (ocol >= OL) columns.
        #pragma unroll
        for (int p = 0; p < NSTEP; ++p) {
            int col = i0 + 4 * p + 2 * half;
            if (col > L_SZ - 2) col = L_SZ - 2;
            const v2f a = *(const v2f*)(xrow + col);
            const v2f b = ldsB[p * 32 + lane];
            acc = __builtin_amdgcn_wmma_f32_16x16x4_f32(
                      false, a, false, b, (short)0, acc, false, false);
        }
    }

    // D layout: VGPR j -> lanes 0-15: M=j, lanes 16-31: M=j+8; N = lane&15.
    const int ocol = i0 + n;
    if (ocol < OL_SZ) {
        #pragma unroll
        for (int j = 0; j < 8; ++j) {
            const int orow = rowBase + j + 8 * half;
            out[(size_t)orow * OL_SZ + ocol] = acc[j];
        }
    }
}

extern "C" void kernel_launch(void* const* d_in, const int* in_sizes, int n_in,
                              void* d_out, int out_size, void* d_ws, size_t ws_size,
                              hipStream_t stream) {
    const float* x   = (const float*)d_in[0];  // [32, 512, 4096] f32
    const float* w   = (const float*)d_in[1];  // [1, 128, 1]     f32
    float*       out = (float*)d_out;          // [32, 512, 3969] f32
    (void)in_sizes; (void)n_in; (void)out_size; (void)d_ws; (void)ws_size;

    const int numColTiles = (OL_SZ + 15) / 16;          // 249
    dim3 grid((numColTiles + 7) / 8, BC_SZ / 16, 1);    // 32 x 1024
    Conv1d_fir_wmma_kernel<<<grid, 256, 0, stream>>>(x, w, out, numColTiles);
}